// OutputLayer_9320079032493
// MI455X (gfx1250) — compile-verified
//
#include <hip/hip_runtime.h>
#include <math.h>
#include <stdint.h>

typedef float v2f __attribute__((ext_vector_type(2)));
typedef float v8f __attribute__((ext_vector_type(8)));
typedef int   v4i __attribute__((vector_size(4 * sizeof(int))));

#define B_     32
#define D_     512
#define T_     2048
#define F_G    (4 * D_)     // 2048 feature rows in G
#define FW     (5 * D_)     // 2560 weights per head
#define KB     64           // K rows staged in LDS per pipeline stage
#define STR    132          // padded LDS row stride (floats)
#define TILEW  128          // t columns per workgroup
#define NTHR   256          // 8 waves
#define ROUNDS ((KB * (TILEW / 4)) / NTHR)   // float4 slots per thread = 8

#if __has_builtin(__builtin_amdgcn_global_load_async_to_lds_b128)
#define HAVE_ASYNC 1
#else
#define HAVE_ASYNC 0
#endif

typedef __attribute__((address_space(1))) v4i* g_v4i_p;
typedef __attribute__((address_space(3))) v4i* l_v4i_p;

// Generic->AS1: bit pattern identical.
__device__ __forceinline__ g_v4i_p to_global(const void* p) {
    return (g_v4i_p)(uintptr_t)p;
}
// Generic->AS3: low 32 bits of a flat LDS-aperture address are the LDS offset
// (ISA 10.2).
__device__ __forceinline__ l_v4i_p to_lds(void* p) {
    return (l_v4i_p)(uintptr_t)(uint32_t)(uintptr_t)p;
}

template <int N>
__device__ __forceinline__ void wait_async() {
#if HAVE_ASYNC
#if __has_builtin(__builtin_amdgcn_s_wait_asynccnt)
    __builtin_amdgcn_s_wait_asynccnt(N);
#else
    asm volatile("s_wait_asynccnt %0" :: "i"(N) : "memory");
#endif
#endif
}

// Issue the ROUNDS async b128 copies that stage a KB x 128 fp32 tile into LDS.
__device__ __forceinline__ void issue_tile(const float* __restrict__ X, int f0,
                                           int t0_block, int tid,
                                           float* __restrict__ buf)
{
#pragma unroll
    for (int r = 0; r < ROUNDS; ++r) {
        const int q   = r * NTHR + tid;     // float4 slot 0..2047
        const int row = q >> 5;             // 32 float4 per 128-col row
        const int c4  = (q & 31) << 2;
        const float* gsrc = X + (size_t)(f0 + row) * T_ + t0_block + c4;
        float*       ldst = buf + row * STR + c4;
#if HAVE_ASYNC
        __builtin_amdgcn_global_load_async_to_lds_b128(to_global(gsrc), to_lds(ldst),
                                                       /*offset=*/0, /*cpol=*/0);
#else
        *(float4*)ldst = *(const float4*)gsrc;
#endif
    }
}

// 16 chained V_WMMA_F32_16X16X4_F32 over one staged tile.
// A row 0 = w1 head, row 1 = w2 head (rest zero); B = 4x16 data tile.
template <bool USE1, bool USE2>
__device__ __forceinline__ void compute_tile(const float* __restrict__ buf,
                                             const float* __restrict__ lw1,
                                             const float* __restrict__ lw2,
                                             int fbase, int lane, int local_t, v8f& c)
{
    const int n    = lane & 15;             // N col index == A row index m
    const int koff = (lane >> 4) << 1;      // lane-half K base (A/B layout)
#pragma unroll
    for (int k = 0; k < KB; k += 4) {
        const int kk = k + koff;
        v2f bv;
        bv.x = buf[(kk    ) * STR + local_t + n];
        bv.y = buf[(kk + 1) * STR + local_t + n];

        float w1a = 0.f, w1b = 0.f, w2a = 0.f, w2b = 0.f;
        if (USE1) { w1a = lw1[fbase + kk]; w1b = lw1[fbase + kk + 1]; }
        if (USE2) { w2a = lw2[fbase + kk]; w2b = lw2[fbase + kk + 1]; }

        v2f av;
        av.x = (n == 0) ? w1a : ((n == 1) ? w2a : 0.f);
        av.y = (n == 0) ? w1b : ((n == 1) ? w2b : 0.f);

        c = __builtin_amdgcn_wmma_f32_16x16x4_f32(
                false, av, false, bv, (short)0, c, false, false);
    }
}

// Double-buffered async pipeline over F rows of one source tensor.
template <bool USE1, bool USE2>
__device__ __forceinline__ void seg_accum(const float* __restrict__ X, int F,
                                          const float* __restrict__ lw1,
                                          const float* __restrict__ lw2,
                                          int t0_block, int tid,
                                          float* __restrict__ buf0,
                                          float* __restrict__ buf1, v8f& c)
{
    const int lane    = tid & 31;
    const int local_t = (tid >> 5) * 16;
    float* bufs[2] = { buf0, buf1 };
    int cur = 0;

    issue_tile(X, 0, t0_block, tid, bufs[0]);           // prime
    for (int f0 = 0; f0 < F; f0 += KB) {
        if (f0 + KB < F) {
            issue_tile(X, f0 + KB, t0_block, tid, bufs[cur ^ 1]);
            wait_async<ROUNDS>();   // retire current tile, keep next in flight
        } else {
            wait_async<0>();        // drain
        }
        __syncthreads();            // tile visible across all waves
        compute_tile<USE1, USE2>(bufs[cur], lw1, lw2, f0, lane, local_t, c);
        __syncthreads();            // all readers done before buffer reuse
        cur ^= 1;
    }
}

__global__ __launch_bounds__(NTHR) void logits_kernel(
    const float* __restrict__ G,  const float* __restrict__ m1,
    const float* __restrict__ m2, const float* __restrict__ wp1,
    const float* __restrict__ wp2,
    float* __restrict__ ws1, float* __restrict__ ws2)
{
    __shared__ float tile[2][KB * STR];
    __shared__ float wsm[2 * FW];
    const int tid      = threadIdx.x;
    const int b        = blockIdx.y;
    const int t0_block = blockIdx.x * TILEW;

    // Preload both weight vectors into LDS once.
    for (int i = tid; i < FW; i += NTHR) {
        wsm[i]      = wp1[i];
        wsm[FW + i] = wp2[i];
    }
    __syncthreads();

    v8f c = {};
    seg_accum<true, true >(G  + (size_t)b * F_G * T_, F_G, wsm,             wsm + FW,
                           t0_block, tid, tile[0], tile[1], c);
    seg_accum<true, false>(m1 + (size_t)b * D_  * T_, D_,  wsm + F_G,       wsm,
                           t0_block, tid, tile[0], tile[1], c);
    seg_accum<false, true>(m2 + (size_t)b * D_  * T_, D_,  wsm,             wsm + FW + F_G,
                           t0_block, tid, tile[0], tile[1], c);

    // C/D layout: VGPR i, lanes 0-15 -> M=i, N=lane. M=0 -> p1, M=1 -> p2.
    const int lane = tid & 31;
    const int t    = t0_block + (tid >> 5) * 16 + (lane & 15);
    if (lane < 16) {
        ws1[(size_t)b * T_ + t] = c[0];
        ws2[(size_t)b * T_ + t] = c[1];
    }
}

__global__ __launch_bounds__(NTHR) void softmax_kernel(
    const float* __restrict__ ws, float* __restrict__ out)
{
    const int bc = blockIdx.x;            // b*2 + ch
    const int b  = bc >> 1, ch = bc & 1;
    const float* row  = ws  + (size_t)ch * B_ * T_ + (size_t)b * T_;
    float*       orow = out + (size_t)bc * T_;
    const int tid = threadIdx.x;

    float v[8];
    float lmax = -INFINITY;
#pragma unroll
    for (int i = 0; i < 8; ++i) {
        v[i] = row[tid + i * NTHR];
        lmax = fmaxf(lmax, v[i]);
    }

    __shared__ float red[NTHR];
    red[tid] = lmax;
    __syncthreads();
    for (int s = NTHR / 2; s > 0; s >>= 1) {
        if (tid < s) red[tid] = fmaxf(red[tid], red[tid + s]);
        __syncthreads();
    }
    const float mx = red[0];
    __syncthreads();

    float lsum = 0.f;
#pragma unroll
    for (int i = 0; i < 8; ++i) {
        v[i] = __expf(v[i] - mx);
        lsum += v[i];
    }
    red[tid] = lsum;
    __syncthreads();
    for (int s = NTHR / 2; s > 0; s >>= 1) {
        if (tid < s) red[tid] += red[tid + s];
        __syncthreads();
    }
    const float inv = 1.f / red[0];

#pragma unroll
    for (int i = 0; i < 8; ++i)
        orow[tid + i * NTHR] = v[i] * inv;
}

extern "C" void kernel_launch(void* const* d_in, const int* in_sizes, int n_in,
                              void* d_out, int out_size, void* d_ws, size_t ws_size,
                              hipStream_t stream) {
    const float* G   = (const float*)d_in[0];
    const float* m1  = (const float*)d_in[1];
    const float* m2  = (const float*)d_in[2];
    const float* wp1 = (const float*)d_in[3];
    const float* wp2 = (const float*)d_in[4];
    float* out = (float*)d_out;

    float* ws1 = (float*)d_ws;                   // [B, T] logits head 1
    float* ws2 = ws1 + (size_t)B_ * T_;          // [B, T] logits head 2

    dim3 grid(T_ / TILEW, B_);
    logits_kernel<<<grid, NTHR, 0, stream>>>(G, m1, m2, wp1, wp2, ws1, ws2);
    softmax_kernel<<<B_ * 2, NTHR, 0, stream>>>(ws1, out);
}